// Attention_64235530879045
// MI455X (gfx1250) — compile-verified
//
#include <hip/hip_runtime.h>

// ---------------------------------------------------------------------------
// Reference collapses exactly:
//   softmax rows sum to 1 and v[b,h,m,d] is constant in m  =>
//   attention output (B,N,C) == spe_agg[b,:] broadcast over n
//   final = spe_agg @ W_proj^T + b_proj, replicated over N positions.
// Kernel 1: y[16x768] = pad(spe_agg) @ W_proj^T via V_WMMA_F32_16X16X4_F32
// Kernel 2: out[b,n,:] = y[b,:] + b_proj  (25.2 MB store-bound, float4)
// ---------------------------------------------------------------------------

typedef __attribute__((ext_vector_type(2))) float v2f;
typedef __attribute__((ext_vector_type(8))) float v8f;

namespace {
constexpr int kB = 8;     // batch
constexpr int kN = 1024;  // sequence
constexpr int kC = 768;   // channels
}

// One wave32 per 16-column output tile: 48 blocks x 32 threads.
// A (16x4 f32): lanes 0-15 = rows M0-15 with K={0,1}; lanes 16-31 same rows, K={2,3}.
// B (4x16 f32): lanes 0-15 = cols N0-15 with K={0,1}; lanes 16-31 same cols, K={2,3}.
// D (16x16 f32): VGPR v holds row M=v (lanes 0-15) / M=v+8 (lanes 16-31), col = lane&15.
__global__ __launch_bounds__(32)
void proj_gemm_wmma(const float* __restrict__ spe,   // (8,768)
                    const float* __restrict__ Wp,    // (768,768) row-major: Wp[out][in]
                    float* __restrict__ y)           // (8,768) workspace
{
  const int j0   = blockIdx.x * 16;       // output column tile
  const int lane = threadIdx.x & 31;
  const int r    = lane & 15;             // M for A-frag, N for B-frag
  const int hi   = lane >> 4;             // K half-select (adds +2 to K)

  // zero-pad rows 8..15 of A without divergence around the WMMA
  const float pad = (r < kB) ? 1.0f : 0.0f;
  const float* aptr = spe + (size_t)(r & 7) * kC + 2 * hi;          // A[m][k + 2*hi + {0,1}]
  const float* bptr = Wp  + (size_t)(j0 + r) * kC + 2 * hi;         // B[k+kk][n] = Wp[j0+n][k+kk]

  v8f acc = {};
#pragma unroll 4
  for (int k = 0; k < kC; k += 4) {
    v2f a, b;
    a.x = aptr[k]     * pad;
    a.y = aptr[k + 1] * pad;
    b.x = bptr[k];
    b.y = bptr[k + 1];
    // (neg_a, A, neg_b, B, c_mod, C, reuse_a, reuse_b)
    acc = __builtin_amdgcn_wmma_f32_16x16x4_f32(false, a, false, b,
                                                (short)0, acc, false, false);
  }

  // Rows 0..7 are the live batches; they live in lanes 0-15 (hi==0), VGPR v = row v.
  if (hi == 0) {
#pragma unroll
    for (int v = 0; v < kB; ++v) {
      y[(size_t)v * kC + j0 + r] = acc[v];
    }
  }
}

// Store-bandwidth-bound broadcast: one float4 (16B) per thread, coalesced along C.
__global__ __launch_bounds__(256)
void bcast_bias(const float* __restrict__ y,      // (8,768)
                const float* __restrict__ bias,   // (768)
                float4* __restrict__ out)         // (8,1024,768) as float4
{
  constexpr int C4 = kC / 4;                      // 192 float4 per row
  const int i  = blockIdx.x * 256 + threadIdx.x;  // global float4 index
  const int c4 = i % C4;
  const int b  = i / (C4 * kN);

  const float4* y4 = (const float4*)y;
  const float4* b4 = (const float4*)bias;
  float4 v  = y4[(size_t)b * C4 + c4];
  float4 bb = b4[c4];
  v.x += bb.x; v.y += bb.y; v.z += bb.z; v.w += bb.w;
  out[i] = v;
}

extern "C" void kernel_launch(void* const* d_in, const int* in_sizes, int n_in,
                              void* d_out, int out_size, void* d_ws, size_t ws_size,
                              hipStream_t stream) {
  (void)in_sizes; (void)n_in; (void)out_size; (void)ws_size;
  // setup_inputs order: x, spe_agg, W_qkv, W_proj, b_proj
  const float* spe = (const float*)d_in[1];
  const float* Wp  = (const float*)d_in[3];
  const float* bp  = (const float*)d_in[4];
  float* y = (float*)d_ws;   // 8*768 floats = 24 KB scratch

  proj_gemm_wmma<<<kC / 16, 32, 0, stream>>>(spe, Wp, y);

  constexpr int total4 = kB * kN * (kC / 4);      // 1,572,864 float4
  bcast_bias<<<total4 / 256, 256, 0, stream>>>(y, bp, (float4*)d_out);
}